// OWAPoolingNew_11879879542554
// MI455X (gfx1250) — compile-verified
//
#include <hip/hip_runtime.h>

// OWA 2x2 pooling, NHWC fp32. x: (16,224,224,128), kernel: (4,1) -> out: (16,112,112,128).
// Pure streaming (zero reuse): memory-bound at ~514 MB => ~22us floor at 23.3 TB/s.
// Design: 1 thread = 1 output pixel x 4 channels (float4). 4x global_load_b128 (NT),
// 5-stage min/max sorting network, 4 FMAs, 1x global_store_b128 (NT).

typedef float v4f __attribute__((ext_vector_type(4)));

#define B_   16
#define H_   224
#define W_   224
#define C_   128
#define PH_  (H_ / 2)   // 112
#define PW_  (W_ / 2)   // 112
#define C4_  (C_ / 4)   // 32 float4 groups per pixel
#define ROW4 (W_ * C4_) // 7168 float4 per input row

__global__ __launch_bounds__(256) void owa_pool2x2_kernel(
    const v4f* __restrict__ x,   // input, float4 view
    const float* __restrict__ k, // 4 OWA weights (desc-rank order), uniform -> s_load
    v4f* __restrict__ out)       // output, float4 view
{
    // grid = (PW_*C4_/256 = 14, PH_ = 112, B_ = 16), block = 256
    const int idx = blockIdx.x * 256 + threadIdx.x; // [0, 3584)
    const int c4  = idx & (C4_ - 1);                // channel group
    const int pw  = idx >> 5;                       // output col
    const int ph  = blockIdx.y;
    const int b   = blockIdx.z;

    // input pixel (b, 2*ph, 2*pw) in float4 units; total < 2^31 bytes so int math is safe
    const int base = ((b * H_ + 2 * ph) * W_ + 2 * pw) * C4_ + c4;

    // 4 window corners: same VGPR address, immediate offsets (512B / 114688B / 115200B
    // all fit the signed 24-bit IOFFSET). Non-temporal: single-use stream, spare L2.
    v4f a0 = __builtin_nontemporal_load(&x[base]);
    v4f a1 = __builtin_nontemporal_load(&x[base + C4_]);
    v4f a2 = __builtin_nontemporal_load(&x[base + ROW4]);
    v4f a3 = __builtin_nontemporal_load(&x[base + ROW4 + C4_]);

    const float k0 = k[0], k1 = k[1], k2 = k[2], k3 = k[3];

    v4f r;
#pragma unroll
    for (int i = 0; i < 4; ++i) {
        const float a = a0[i], bb = a1[i], c = a2[i], d = a3[i];
        // 4-element sorting network (descending), 5 compare-exchanges:
        const float t0 = fmaxf(a, bb), t1 = fminf(a, bb);
        const float t2 = fmaxf(c, d),  t3 = fminf(c, d);
        const float s0 = fmaxf(t0, t2), u  = fminf(t0, t2);
        const float v  = fmaxf(t1, t3), s3 = fminf(t1, t3);
        const float s1 = fmaxf(u, v),   s2 = fminf(u, v);
        // learned ordered-weighted sum (k0 on the max)
        r[i] = fmaf(k0, s0, fmaf(k1, s1, fmaf(k2, s2, k3 * s3)));
    }

    const int o = ((b * PH_ + ph) * PW_ + pw) * C4_ + c4;
    __builtin_nontemporal_store(r, &out[o]);
}

extern "C" void kernel_launch(void* const* d_in, const int* in_sizes, int n_in,
                              void* d_out, int out_size, void* d_ws, size_t ws_size,
                              hipStream_t stream)
{
    (void)in_sizes; (void)n_in; (void)out_size; (void)d_ws; (void)ws_size;
    const v4f*   x = (const v4f*)d_in[0];
    const float* k = (const float*)d_in[1];
    v4f*       out = (v4f*)d_out;

    dim3 block(256, 1, 1);
    dim3 grid(PW_ * C4_ / 256, PH_, B_); // (14, 112, 16)
    hipLaunchKernelGGL(owa_pool2x2_kernel, grid, block, 0, stream, x, k, out);
}